// IntVOS_27015344292341
// MI455X (gfx1250) — compile-verified
//
#include <hip/hip_runtime.h>

// Shapes are compile-time constants from the reference.
#define H    128
#define W    128
#define C    100
#define NOBJ 3
#define MD   15              // max_distance
#define KSTEPS (C / 4)       // 25 WMMA K-steps of 4 for V_WMMA_F32_16X16X4_F32
#define INFV 1e30f
#define ROWF (W * C)         // floats per image row = 12800
#define ROWB (ROWF * 4)      // bytes per image row  = 51200

typedef __attribute__((ext_vector_type(2))) float v2f;
typedef __attribute__((ext_vector_type(8))) float v8f;
typedef unsigned int u32;
typedef __attribute__((ext_vector_type(4))) u32 v4u;
typedef __attribute__((ext_vector_type(8))) int  v8i;
typedef __attribute__((ext_vector_type(4))) int  v4i;

#if defined(__has_builtin)
#if __has_builtin(__builtin_amdgcn_tensor_load_to_lds) && \
    __has_builtin(__builtin_amdgcn_s_wait_tensorcnt)
#define USE_TDM 1
#endif
#endif

// ---------------------------------------------------------------------------
// Pass 1: per-pixel squared norms of both embeddings into d_ws.
// ---------------------------------------------------------------------------
__global__ __launch_bounds__(256) void norms_kernel(
    const float* __restrict__ prevEmb,
    const float* __restrict__ queryEmb,
    float* __restrict__ pnorm,   // [H*W]
    float* __restrict__ qnorm)   // [H*W]
{
    const int p = blockIdx.x * blockDim.x + threadIdx.x;
    if (p >= H * W) return;
    const float* a = prevEmb  + (size_t)p * C;   // 400B stride -> 16B aligned
    const float* b = queryEmb + (size_t)p * C;
    float sa = 0.f, sb = 0.f;
#pragma unroll
    for (int k = 0; k < C; k += 4) {
        float4 va = *(const float4*)(a + k);
        float4 vb = *(const float4*)(b + k);
        sa += va.x * va.x + va.y * va.y + va.z * va.z + va.w * va.w;
        sb += vb.x * vb.x + vb.y * vb.y + vb.z * vb.z + vb.w * vb.w;
    }
    pnorm[p] = sa;
    qnorm[p] = sb;
}

#ifdef USE_TDM
// Issue a TDM copy of one contiguous image row (12800 f32) into LDS.
// D# per CDNA5 ISA ch.8: group0 = {count/flags, lds_addr, global_addr, type},
// group1 = dims/strides for a 1-D tile, groups 2/3 unused (<=2D tensor).
// This toolchain declares the 6-arg builtin:
//   (uint32x4 g0, int32x8 g1, int32x4, int32x4, int32x8, i32 cpol)
__device__ __forceinline__ void tdm_load_row(const float* gsrc, u32 ldsByteOff)
{
    unsigned long long ga = (unsigned long long)(uintptr_t)gsrc;
    v4u g0;
    g0.x = 1u;                                   // count=1, user descriptor
    g0.y = ldsByteOff;                           // lds_addr (bytes)
    g0.z = (u32)(ga & 0xFFFFFFFFu);              // global_addr[31:0]
    g0.w = (u32)((ga >> 32) & 0x01FFFFFFu)       // global_addr[56:32]
         | (2u << 30);                           // type=2 ("image")
    v8i g1;
    g1[0] = (int)(2u << 16);                     // data_size=2 (4 bytes)
    g1[1] = (int)(((u32)ROWF & 0xFFFFu) << 16);  // tensor_dim0[15:0]
    g1[2] = (int)(((u32)ROWF >> 16)              // tensor_dim0[31:16]
         | (1u << 16));                          // tensor_dim1 = 1
    g1[3] = (int)(((u32)ROWF & 0xFFFFu) << 16);  // tile_dim0 = 12800
    g1[4] = 0;                                   // tile_dim1/2 unused
    g1[5] = (int)ROWF;                           // tensor_dim0_stride[31:0]
    g1[6] = 0;
    g1[7] = 0;
    v4i z4 = {0, 0, 0, 0};
    v8i z8 = {0, 0, 0, 0, 0, 0, 0, 0};
    __builtin_amdgcn_tensor_load_to_lds(g0, g1, z4, z4, z8, 0);
}
#endif

// Masked running-min update without fminf canonicalization:
// mn = (take && d < mn) ? d : mn
__device__ __forceinline__ void min_upd(float& mn, bool take, float d)
{
    if (take & (d < mn)) mn = d;
}

// ---------------------------------------------------------------------------
// Pass 2: one workgroup (8 wave32) per query row; wave w owns 16-col strip w.
// Reference rows staged through LDS by the Tensor Data Mover (double buffer).
// Cross terms via exact-fp32 V_WMMA_F32_16X16X4_F32.
// ---------------------------------------------------------------------------
extern __shared__ float ldsrow[];   // 2 * ROWF floats (TDM) / ROWF (fallback)

__global__ __launch_bounds__(256) void local_nn_wmma_kernel(
    const float* __restrict__ prevEmb,   // [H, W, C]
    const float* __restrict__ queryEmb,  // [H, W, C]
    const int*   __restrict__ labels,    // [H, W]
    const int*   __restrict__ gtIds,     // [NOBJ]
    const float* __restrict__ pnorm,     // [H*W]
    const float* __restrict__ qnorm,     // [H*W]
    float*       __restrict__ out)       // [H, W, NOBJ]
{
    const int qrow  = blockIdx.x;
    const int wave  = threadIdx.x >> 5;
    const int qbase = wave << 4;
    const int lane  = threadIdx.x & 31;
    const int nl    = lane & 15;
    const int hl    = lane >> 4;

    // ---- A-matrix cache: query strip, all 25 K-steps (registers) ----
    const float* qptr = queryEmb + (size_t)(qrow * W + qbase + nl) * C + hl * 2;
    v2f acache[KSTEPS];
#pragma unroll
    for (int kk = 0; kk < KSTEPS; ++kk)
        acache[kk] = *(const v2f*)(qptr + kk * 4);

    // ||q||^2 for the 8 D-rows (M = v + 8*hl) this lane covers
    const float qn_full = qnorm[qrow * W + qbase + nl];
    float qn_v[8];
#pragma unroll
    for (int v = 0; v < 8; ++v)
        qn_v[v] = __shfl(qn_full, v + 8 * hl, 32);

    const int g0 = gtIds[0];
    const int g1 = gtIds[1];
    const int g2 = gtIds[2];

    float mn[8][NOBJ];
#pragma unroll
    for (int v = 0; v < 8; ++v)
#pragma unroll
        for (int o = 0; o < NOBJ; ++o)
            mn[v][o] = 1.0f;            // reference fill for masked-out offsets

    const int r0 = (qrow - MD < 0) ? 0 : qrow - MD;
    const int r1 = (qrow + MD > H - 1) ? H - 1 : qrow + MD;

#ifdef USE_TDM
    int cur = 0;
    if (wave == 0) tdm_load_row(prevEmb + (size_t)r0 * ROWF, 0u);
#endif

    for (int r = r0; r <= r1; ++r) {
#ifdef USE_TDM
        if (wave == 0) __builtin_amdgcn_s_wait_tensorcnt(0);
        __syncthreads();                            // row r staged; old readers done
        if (wave == 0 && r + 1 <= r1)
            tdm_load_row(prevEmb + (size_t)(r + 1) * ROWF,
                         (u32)((cur ^ 1) * ROWB)); // fill other buffer
        const float* rowbuf = ldsrow + cur * ROWF;
#else
        __syncthreads();                            // previous row's readers done
        {   // cooperative copy of row r (3200 float4 by 256 threads)
            const float4* src = (const float4*)(prevEmb + (size_t)r * ROWF);
            float4*       dst = (float4*)ldsrow;
            for (int i = threadIdx.x; i < ROWF / 4; i += 256) dst[i] = src[i];
        }
        __syncthreads();
        const float* rowbuf = ldsrow;
#endif

#pragma unroll
        for (int t = 0; t < 3; ++t) {               // t literal -> specialized
            const int cbase = qbase + (t - 1) * 16; // uniform across wave
            if (cbase < 0 || cbase >= W) continue;  // fully out-of-frame tile

            const int refcol = cbase + nl;          // always in [0, W)

            const float pn  = pnorm[r * W + refcol];
            const int   lab = labels[r * W + refcol];

            // B fragments from the staged LDS row (ds_load_b64 clause)
            const float* pptr = rowbuf + refcol * C + hl * 2;
            v2f bfrags[KSTEPS];
#pragma unroll
            for (int kk = 0; kk < KSTEPS; ++kk)
                bfrags[kk] = *(const v2f*)(pptr + kk * 4);

            v8f acc = {0.f, 0.f, 0.f, 0.f, 0.f, 0.f, 0.f, 0.f};
#pragma unroll
            for (int kk = 0; kk < KSTEPS; ++kk)
                acc = __builtin_amdgcn_wmma_f32_16x16x4_f32(
                        false, acache[kk], false, bfrags[kk],
                        (short)0, acc, false, false);

            const bool m0 = (lab == g0);
            const bool m1 = (lab == g1);
            const bool m2 = (lab == g2);

            // Column-window mask with 16-aligned tiles:
            //   t==1 (center): always valid; t==0: m < n; t==2: m > n
#pragma unroll
            for (int v = 0; v < 8; ++v) {
                const int m = v + 8 * hl;
                bool cok = true;
                if (t == 0) cok = (m < nl);
                if (t == 2) cok = (m > nl);
                const float d = __builtin_fmaf(-2.0f, acc[v], qn_v[v] + pn);
                min_upd(mn[v][0], cok & m0, d);
                min_upd(mn[v][1], cok & m1, d);
                min_upd(mn[v][2], cok & m2, d);
            }
        }
#ifdef USE_TDM
        cur ^= 1;
#endif
    }

    // ---- reduce mins across the 16 N-lanes of each half ----
#pragma unroll
    for (int v = 0; v < 8; ++v) {
#pragma unroll
        for (int o = 0; o < NOBJ; ++o) {
            float m = mn[v][o];
            float s;
            s = __shfl_xor(m, 1, 32); if (s < m) m = s;
            s = __shfl_xor(m, 2, 32); if (s < m) m = s;
            s = __shfl_xor(m, 4, 32); if (s < m) m = s;
            s = __shfl_xor(m, 8, 32); if (s < m) m = s;
            if (nl == 0) {
                const int col = qbase + v + 8 * hl;
                out[(size_t)(qrow * W + col) * NOBJ + o] = m;
            }
        }
    }
}

extern "C" void kernel_launch(void* const* d_in, const int* in_sizes, int n_in,
                              void* d_out, int out_size, void* d_ws, size_t ws_size,
                              hipStream_t stream) {
    const float* prevEmb  = (const float*)d_in[0];
    const float* queryEmb = (const float*)d_in[1];
    const int*   labels   = (const int*)d_in[2];
    const int*   gtIds    = (const int*)d_in[3];
    // d_in[4] = max_distance scalar (compile-time MD=15)

    float* pnorm = (float*)d_ws;             // [H*W]
    float* qnorm = pnorm + H * W;            // [H*W]  (128 KB of ws)
    float* out   = (float*)d_out;

    norms_kernel<<<dim3((H * W + 255) / 256), dim3(256), 0, stream>>>(
        prevEmb, queryEmb, pnorm, qnorm);

    // One workgroup per query row; 2 * 51200 B dynamic LDS for the
    // double-buffered TDM row staging (within the 320 KB WGP LDS).
    local_nn_wmma_kernel<<<dim3(H), dim3(256), 2 * ROWB, stream>>>(
        prevEmb, queryEmb, labels, gtIds, pnorm, qnorm, out);
}